// SummaRuNNer_11759620457007
// MI455X (gfx1250) — compile-verified
//
#include <hip/hip_runtime.h>
#include <math.h>

typedef __attribute__((ext_vector_type(2))) float v2f;
typedef __attribute__((ext_vector_type(8))) float v8f;

#define SS  128   // sentences
#define TT  100   // tokens per sentence
#define EE  200   // embedding dim
#define HH  200   // GRU hidden
#define FCD 100   // fc out dim
#define PDD 50    // pos emb dim

// ---------------------------------------------------------------------------
// WMMA GEMM:  C = A(MxK) @ op(B) + bias.  op(B)=B^T when TRANSB (B stored NxK),
// else B stored (K,N).  Exact f32 via V_WMMA_F32_16X16X4_F32 (wave32).
// Each wave computes a 16x64 strip (4 accumulators); fragment loads are
// software-pipelined (double-buffered) so the next K-step's 5 global_load_b64
// are in flight while the current step's 4 WMMAs execute.  gridDim.z selects
// between two (A,B,bias,C) sets so fwd/rev GRU directions share one launch.
// Edge tiles clamp load indices (EXEC stays all-1s) and predicate stores.
// ---------------------------------------------------------------------------
template <int TRANSB>
__global__ __launch_bounds__(256) void wmma_gemm_kernel(
    const float* __restrict__ A0, const float* __restrict__ B0,
    const float* __restrict__ bias0, float* __restrict__ C0,
    const float* __restrict__ A1, const float* __restrict__ B1,
    const float* __restrict__ bias1, float* __restrict__ C1,
    int M, int N, int K, int act)
{
  constexpr int TN = 4;                       // 4 x 16 columns per wave
  const float* A    = blockIdx.z ? A1 : A0;
  const float* B    = blockIdx.z ? B1 : B0;
  const float* bias = blockIdx.z ? bias1 : bias0;
  float*       C    = blockIdx.z ? C1 : C0;

  int ntN  = (N + 15) >> 4;
  int ntM  = (M + 15) >> 4;
  int ntNg = (ntN + TN - 1) / TN;
  int wave = (int)blockIdx.x * ((int)blockDim.x >> 5) + ((int)threadIdx.x >> 5);
  if (wave >= ntM * ntNg) return;             // whole-wave uniform exit
  int tm = wave / ntNg, tg = wave - tm * ntNg;
  int m0 = tm << 4;
  int n0 = (tg * TN) << 4;

  int lane = (int)threadIdx.x & 31;
  int half = lane >> 4;                       // 0: K=k..k+1, 1: K=k+2..k+3
  int l    = lane & 15;
  int arow = m0 + l; if (arow >= M) arow = M - 1;   // clamp, EXEC stays full

  int bcol[TN];
#pragma unroll
  for (int i = 0; i < TN; ++i) {
    int c = n0 + i * 16 + l;
    bcol[i] = (c < N) ? c : (N - 1);
  }

  const v2f* Arow = (const v2f*)(A + (long)arow * K);
  const v2f* Brow[TN];
  if (TRANSB) {
#pragma unroll
    for (int i = 0; i < TN; ++i) Brow[i] = (const v2f*)(B + (long)bcol[i] * K);
  }

  auto loadB = [&](int i, int k) -> v2f {
    if (TRANSB) {
      return Brow[i][(k >> 1) + half];
    } else {
      int kk = k + (half << 1);
      v2f b;
      b.x = B[(long)kk * N + bcol[i]];
      b.y = B[(long)(kk + 1) * N + bcol[i]];
      return b;
    }
  };

  v8f acc[TN] = {};

  // ---- software-pipelined K loop: prefetch (k+4) while computing k ----
  v2f a_cur = Arow[half];
  v2f b_cur[TN];
#pragma unroll
  for (int i = 0; i < TN; ++i) b_cur[i] = loadB(i, 0);

  for (int k = 0; k < K - 4; k += 4) {
    v2f a_nxt = Arow[((k + 4) >> 1) + half];
    v2f b_nxt[TN];
#pragma unroll
    for (int i = 0; i < TN; ++i) b_nxt[i] = loadB(i, k + 4);
#pragma unroll
    for (int i = 0; i < TN; ++i)
      acc[i] = __builtin_amdgcn_wmma_f32_16x16x4_f32(false, a_cur, false, b_cur[i],
                                                     (short)0, acc[i], false, false);
    a_cur = a_nxt;
#pragma unroll
    for (int i = 0; i < TN; ++i) b_cur[i] = b_nxt[i];
  }
#pragma unroll
  for (int i = 0; i < TN; ++i)
    acc[i] = __builtin_amdgcn_wmma_f32_16x16x4_f32(false, a_cur, false, b_cur[i],
                                                   (short)0, acc[i], false, false);

  // ---- epilogue: bias + optional tanh, predicated stores ----
#pragma unroll
  for (int i = 0; i < TN; ++i) {
    int ccol = n0 + i * 16 + l;
    if (ccol < N) {
      float bv = bias ? bias[ccol] : 0.0f;
#pragma unroll
      for (int r = 0; r < 8; ++r) {
        int crow = m0 + r + (half << 3);
        if (crow < M) {
          float v = acc[i][r] + bv;
          if (act == 1) v = tanhf(v);
          C[(long)crow * N + ccol] = v;
        }
      }
    }
  }
}

// lengths[s] = sum_t sign(x[s,t])
__global__ void lengths_kernel(const int* __restrict__ x, float* __restrict__ len)
{
  int s = blockIdx.x * blockDim.x + threadIdx.x;
  if (s >= SS) return;
  int c = 0;
  for (int t = 0; t < TT; ++t) {
    int v = x[s * TT + t];
    c += (v > 0) ? 1 : ((v < 0) ? -1 : 0);
  }
  len[s] = (float)c;
}

// word_feat[(s*T+t), e] = emb[x[s,t], e]
__global__ void gather_kernel(const int* __restrict__ x, const float* __restrict__ emb,
                              float* __restrict__ wf, long total)
{
  long idx = (long)blockIdx.x * blockDim.x + threadIdx.x;
  if (idx >= total) return;
  long r = idx / EE;
  int  e = (int)(idx - r * EE);
  wf[idx] = emb[(long)x[r] * EE + e];
}

// Fused GRU gate update for both directions (blockIdx.z = dir).
// gi pointers pre-offset to timestep t; out pointers pre-offset to (t, colOff).
__global__ void gru_gate_kernel(
    const float* __restrict__ giF, const float* __restrict__ giR, long giRowStride,
    const float* __restrict__ ghF, const float* __restrict__ ghR,
    float* __restrict__ hF, float* __restrict__ hR,
    float* __restrict__ outF, float* __restrict__ outR, long outRowStride,
    int B, int H)
{
  int idx = blockIdx.x * blockDim.x + threadIdx.x;
  if (idx >= B * H) return;
  int s = idx / H, j = idx - s * H;
  const float* gi = blockIdx.z ? giR : giF;
  const float* gh = blockIdx.z ? ghR : ghF;
  float*       h  = blockIdx.z ? hR  : hF;
  float*       o  = blockIdx.z ? outR : outF;

  float ir = gi[(long)s * giRowStride + j];
  float iz = gi[(long)s * giRowStride + H + j];
  float in = gi[(long)s * giRowStride + 2 * H + j];
  float hr = gh[(long)s * 3 * H + j];
  float hz = gh[(long)s * 3 * H + H + j];
  float hn = gh[(long)s * 3 * H + 2 * H + j];
  float r = 1.0f / (1.0f + expf(-(ir + hr)));
  float z = 1.0f / (1.0f + expf(-(iz + hz)));
  float n = tanhf(in + r * hn);
  float hold = h[(long)s * H + j];
  float hnew = (1.0f - z) * n + z * hold;
  h[(long)s * H + j] = hnew;
  o[(long)s * outRowStride + j] = hnew;
}

// sent_feat[s,j] = sum_{t<len[s]} word_out[s,t,j] / len[s]      (j in [0,2H))
__global__ void sentfeat_kernel(const float* __restrict__ word_out,
                                const float* __restrict__ len,
                                float* __restrict__ sf)
{
  int idx = blockIdx.x * blockDim.x + threadIdx.x;
  const int D = 2 * HH;
  if (idx >= SS * D) return;
  int s = idx / D, j = idx - s * D;
  int L = (int)len[s];
  float acc = 0.0f;
  for (int t = 0; t < TT; ++t)
    if (t < L) acc += word_out[((long)s * TT + t) * D + j];
  sf[idx] = acc / len[s];
}

// mean over sentences of sent_out -> (2H,)
__global__ void meansent_kernel(const float* __restrict__ so, float* __restrict__ ms)
{
  int j = blockIdx.x * blockDim.x + threadIdx.x;
  const int D = 2 * HH;
  if (j >= D) return;
  float acc = 0.0f;
  for (int s = 0; s < SS; ++s) acc += so[(long)s * D + j];
  ms[j] = acc / (float)SS;
}

// base[s] = h[s]·Wc + h[s]·(Ws@doc) + pos_emb[s,:PD]·Wp + bias
__global__ void base_kernel(const float* __restrict__ h, const float* __restrict__ v,
                            const float* __restrict__ Wc, const float* __restrict__ pos_emb,
                            const float* __restrict__ Wp, const float* __restrict__ bias,
                            float* __restrict__ base)
{
  int s = threadIdx.x;
  if (s >= SS) return;
  float acc = bias[0];
  for (int j = 0; j < FCD; ++j) acc += h[(long)s * FCD + j] * (Wc[j] + v[j]);
  for (int j = 0; j < PDD; ++j) acc += pos_emb[(long)s * PDD + j] * Wp[j];
  base[s] = acc;
}

// sequential sigmoid scan (one block of 128 threads)
__global__ void scan_kernel(const float* __restrict__ base, const float* __restrict__ h,
                            const float* __restrict__ hWr, float* __restrict__ probs)
{
  __shared__ float st[128];
  __shared__ float red[128];
  __shared__ float prob_sh;
  int j = threadIdx.x;
  st[j] = 0.0f;
  __syncthreads();
  for (int t = 0; t < SS; ++t) {
    float val = (j < FCD) ? hWr[(long)t * FCD + j] * tanhf(st[j]) : 0.0f;
    red[j] = val;
    __syncthreads();
    for (int off = 64; off > 0; off >>= 1) {
      if (j < off) red[j] += red[j + off];
      __syncthreads();
    }
    if (j == 0) prob_sh = 1.0f / (1.0f + expf(-(base[t] - red[0])));
    __syncthreads();
    float p = prob_sh;
    if (j < FCD) st[j] += h[(long)t * FCD + j] * p;
    if (j == 0) probs[t] = p;
    __syncthreads();
  }
}

// ---------------------------------------------------------------------------
static void launch_gemm(hipStream_t stream,
                        const float* A0, const float* B0, const float* b0, float* C0,
                        const float* A1, const float* B1, const float* b1, float* C1,
                        int M, int N, int K, int transB, int act, int dual)
{
  const int TN = 4;
  int ntM = (M + 15) / 16, ntN = (N + 15) / 16;
  int ntNg = (ntN + TN - 1) / TN;
  int tiles = ntM * ntNg;
  int blocks = (tiles + 7) / 8;               // 8 waves per 256-thread block
  dim3 grid(blocks, 1, dual ? 2 : 1);
  if (transB)
    wmma_gemm_kernel<1><<<grid, 256, 0, stream>>>(A0, B0, b0, C0, A1, B1, b1, C1,
                                                  M, N, K, act);
  else
    wmma_gemm_kernel<0><<<grid, 256, 0, stream>>>(A0, B0, b0, C0, A1, B1, b1, C1,
                                                  M, N, K, act);
}

extern "C" void kernel_launch(void* const* d_in, const int* in_sizes, int n_in,
                              void* d_out, int out_size, void* d_ws, size_t ws_size,
                              hipStream_t stream)
{
  (void)in_sizes; (void)n_in; (void)out_size; (void)ws_size;
  const int*   x       = (const int*)  d_in[0];
  const float* emb     = (const float*)d_in[1];
  const float* pos_emb = (const float*)d_in[2];
  const float* wWih_f  = (const float*)d_in[3];
  const float* wWhh_f  = (const float*)d_in[4];
  const float* wbih_f  = (const float*)d_in[5];
  const float* wbhh_f  = (const float*)d_in[6];
  const float* wWih_r  = (const float*)d_in[7];
  const float* wWhh_r  = (const float*)d_in[8];
  const float* wbih_r  = (const float*)d_in[9];
  const float* wbhh_r  = (const float*)d_in[10];
  const float* sWih_f  = (const float*)d_in[11];
  const float* sWhh_f  = (const float*)d_in[12];
  const float* sbih_f  = (const float*)d_in[13];
  const float* sbhh_f  = (const float*)d_in[14];
  const float* sWih_r  = (const float*)d_in[15];
  const float* sWhh_r  = (const float*)d_in[16];
  const float* sbih_r  = (const float*)d_in[17];
  const float* sbhh_r  = (const float*)d_in[18];
  const float* fc1_W   = (const float*)d_in[19];
  const float* fc1_b   = (const float*)d_in[20];
  const float* fc2_W   = (const float*)d_in[21];
  const float* fc2_b   = (const float*)d_in[22];
  const float* Wc      = (const float*)d_in[23];
  const float* Ws      = (const float*)d_in[24];
  const float* Wr      = (const float*)d_in[25];
  const float* Wp      = (const float*)d_in[26];
  const float* biasp   = (const float*)d_in[27];
  float* probs = (float*)d_out;

  // --- workspace bump allocator (256B aligned) ---
  size_t off = 0;
  auto alloc = [&](size_t nfloats) -> float* {
    float* p = (float*)((char*)d_ws + off);
    off += ((nfloats * sizeof(float)) + 255) & ~(size_t)255;
    return p;
  };
  float* word_feat = alloc((size_t)SS * TT * EE);        // 12800 x 200
  float* gi_f      = alloc((size_t)SS * TT * 3 * HH);    // 12800 x 600
  float* gi_r      = alloc((size_t)SS * TT * 3 * HH);
  float* word_out  = alloc((size_t)SS * TT * 2 * HH);    // 12800 x 400
  float* h_f       = alloc((size_t)SS * HH);
  float* h_r       = alloc((size_t)SS * HH);
  float* gh_f      = alloc((size_t)SS * 3 * HH);
  float* gh_r      = alloc((size_t)SS * 3 * HH);
  float* lengths   = alloc(SS);
  float* sent_feat = alloc((size_t)SS * 2 * HH);         // 128 x 400
  float* sgi_f     = alloc((size_t)SS * 3 * HH);         // 128 x 600
  float* sgi_r     = alloc((size_t)SS * 3 * HH);
  float* sent_out  = alloc((size_t)SS * 2 * HH);         // 128 x 400
  float* sh_f      = alloc(HH);
  float* sh_r      = alloc(HH);
  float* sgh_f     = alloc(3 * HH);
  float* sgh_r     = alloc(3 * HH);
  float* mean_sent = alloc(2 * HH);
  float* docv      = alloc(FCD);
  float* hmat      = alloc((size_t)SS * FCD);
  float* vvec      = alloc(FCD);
  float* basev     = alloc(SS);
  float* hWr       = alloc((size_t)SS * FCD);

  // --- 1. lengths + embedding gather ---
  lengths_kernel<<<1, 128, 0, stream>>>(x, lengths);
  long gtotal = (long)SS * TT * EE;
  gather_kernel<<<(unsigned)((gtotal + 255) / 256), 256, 0, stream>>>(x, emb, word_feat, gtotal);

  // --- 2. word-level input projections (both dirs, one launch): (12800,200)@(200,600) ---
  launch_gemm(stream, word_feat, wWih_f, wbih_f, gi_f,
                      word_feat, wWih_r, wbih_r, gi_r,
              SS * TT, 3 * HH, EE, /*transB=*/1, /*act=*/0, /*dual=*/1);

  // --- 3. word-level BiGRU recurrence (fwd t=i, rev t=T-1-i, fused per step) ---
  hipMemsetAsync(h_f, 0, (size_t)SS * HH * sizeof(float), stream);
  hipMemsetAsync(h_r, 0, (size_t)SS * HH * sizeof(float), stream);
  for (int i = 0; i < TT; ++i) {
    int tf = i, tr = TT - 1 - i;
    launch_gemm(stream, h_f, wWhh_f, wbhh_f, gh_f,
                        h_r, wWhh_r, wbhh_r, gh_r,
                SS, 3 * HH, HH, 1, 0, 1);
    dim3 ggrid((SS * HH + 255) / 256, 1, 2);
    gru_gate_kernel<<<ggrid, 256, 0, stream>>>(
        gi_f + (long)tf * 3 * HH, gi_r + (long)tr * 3 * HH, (long)TT * 3 * HH,
        gh_f, gh_r, h_f, h_r,
        word_out + (long)tf * 2 * HH, word_out + (long)tr * 2 * HH + HH, (long)TT * 2 * HH,
        SS, HH);
  }

  // --- 4. masked mean over tokens -> sent_feat (128,400) ---
  sentfeat_kernel<<<(SS * 2 * HH + 255) / 256, 256, 0, stream>>>(word_out, lengths, sent_feat);

  // --- 5. sentence-level input projections: (128,400)@(400,600) both dirs ---
  launch_gemm(stream, sent_feat, sWih_f, sbih_f, sgi_f,
                      sent_feat, sWih_r, sbih_r, sgi_r,
              SS, 3 * HH, 2 * HH, 1, 0, 1);

  // --- 6. sentence-level BiGRU recurrence, batch=1, 128 steps ---
  hipMemsetAsync(sh_f, 0, HH * sizeof(float), stream);
  hipMemsetAsync(sh_r, 0, HH * sizeof(float), stream);
  for (int i = 0; i < SS; ++i) {
    int tf = i, tr = SS - 1 - i;
    launch_gemm(stream, sh_f, sWhh_f, sbhh_f, sgh_f,
                        sh_r, sWhh_r, sbhh_r, sgh_r,
                1, 3 * HH, HH, 1, 0, 1);
    dim3 ggrid(1, 1, 2);   // 1*HH = 200 threads needed
    gru_gate_kernel<<<ggrid, 256, 0, stream>>>(
        sgi_f + (long)tf * 3 * HH, sgi_r + (long)tr * 3 * HH, 0,
        sgh_f, sgh_r, sh_f, sh_r,
        sent_out + (long)tf * 2 * HH, sent_out + (long)tr * 2 * HH + HH, 0,
        1, HH);
  }

  // --- 7. doc = tanh(mean(sent_out,0) @ fc1_W.T + fc1_b) ---
  meansent_kernel<<<(2 * HH + 255) / 256, 256, 0, stream>>>(sent_out, mean_sent);
  launch_gemm(stream, mean_sent, fc1_W, fc1_b, docv, 0, 0, 0, 0,
              1, FCD, 2 * HH, /*transB=*/1, /*tanh*/1, 0);   // K = 400

  // --- 8. h = tanh(sent_out @ fc2_W.T + fc2_b): (128,400)->(128,100) ---
  launch_gemm(stream, sent_out, fc2_W, fc2_b, hmat, 0, 0, 0, 0,
              SS, FCD, 2 * HH, 1, 1, 0);

  // --- 9. v = Ws @ doc  (100x100 @ 100 -> 100); base; hWr = h @ Wr ---
  launch_gemm(stream, Ws, docv, nullptr, vvec, 0, 0, 0, 0,
              FCD, 1, FCD, /*transB=*/1, 0, 0);
  base_kernel<<<1, 128, 0, stream>>>(hmat, vvec, Wc, pos_emb, Wp, biasp, basev);
  launch_gemm(stream, hmat, Wr, nullptr, hWr, 0, 0, 0, 0,
              SS, FCD, FCD, /*transB=*/0, 0, 0);

  // --- 10. sequential sigmoid scan -> probs (128,) ---
  scan_kernel<<<1, 128, 0, stream>>>(basev, hmat, hWr, probs);
}